// CTCLayer_71313636983467
// MI455X (gfx1250) — compile-verified
//
#include <hip/hip_runtime.h>
#include <hip/hip_bf16.h>
#include <stdint.h>

// Problem constants (match the reference)
#define B_     1024
#define T_     1024
#define C_     96
#define L_     32
#define S_     65          // 2*L+1 extended states
#define BLANK_ 95          // C-1
#define NEGF   (-1e30f)

// ---------------------------------------------------------------------------
// CDNA5 async global->LDS staging (ASYNCcnt-tracked), inline asm since the
// async-load builtins are not probe-confirmed on this toolchain.
// The instruction offset is added to both the LDS address and the global (SGPR base + VGPR) address.
// So one (ldsoff, goff) pair + offset:0/128/256 covers classes c, c+32, c+64.
// ---------------------------------------------------------------------------
__device__ __forceinline__ void async_load3(unsigned ldsoff, unsigned goff,
                                            unsigned long long gbase) {
  asm volatile(
      "global_load_async_to_lds_b32 %0, %1, %2\n\t"
      "global_load_async_to_lds_b32 %0, %1, %2 offset:128\n\t"
      "global_load_async_to_lds_b32 %0, %1, %2 offset:256"
      :: "v"(ldsoff), "v"(goff), "s"(gbase)
      : "memory");
}
__device__ __forceinline__ void wait_async_le3() {
  asm volatile("s_wait_asynccnt 0x3" ::: "memory");
}
__device__ __forceinline__ void wait_async_0() {
  asm volatile("s_wait_asynccnt 0x0" ::: "memory");
}

// wave32 butterfly reductions (all lanes end with the result)
__device__ __forceinline__ float wave_max32(float v) {
#pragma unroll
  for (int off = 16; off > 0; off >>= 1)
    v = fmaxf(v, __shfl_xor(v, off, 32));
  return v;
}
__device__ __forceinline__ float wave_sum32(float v) {
#pragma unroll
  for (int off = 16; off > 0; off >>= 1)
    v += __shfl_xor(v, off, 32);
  return v;
}

// 3-way log-sum-exp (same numerics as the reference step)
__device__ __forceinline__ float lse3(float x, float y, float z) {
  float m = fmaxf(fmaxf(x, y), z);
  return m + __logf(__expf(x - m) + __expf(y - m) + __expf(z - m));
}

// log-softmax normalizer over the 96 staged logits of one time slice
__device__ __forceinline__ float lse_row(const float* bt, int lane) {
  float x0 = bt[lane], x1 = bt[lane + 32], x2 = bt[lane + 64];
  float m = wave_max32(fmaxf(fmaxf(x0, x1), x2));
  float s = wave_sum32(__expf(x0 - m) + __expf(x1 - m) + __expf(x2 - m));
  return m + __logf(s);
}

// One wave (32 lanes) per batch row. Lane owns extended states s=lane and
// s=lane+32; lane 0 additionally owns s=64. Alpha lives in registers; the
// only LDS traffic is the double-buffered logits slice filled by the async DMA.
__global__ __launch_bounds__(32)
void ctc_fwd_kernel(const int* __restrict__ y_true,
                    const float* __restrict__ y_pred,
                    float* __restrict__ loss) {
  __shared__ float buf[2 * C_];   // double buffer: 96 logits per time step

  const int b    = blockIdx.x;
  const int lane = threadIdx.x;

  const unsigned long long gbase =
      (unsigned long long)(uintptr_t)(y_pred + (size_t)b * T_ * C_);
  const int* yt = y_true + b * L_;

  // LDS byte offsets of the two buffer halves (flat addr [31:0] == LDS offset)
  const unsigned lds0 = (unsigned)(uintptr_t)(buf)      + (unsigned)lane * 4u;
  const unsigned lds1 = (unsigned)(uintptr_t)(buf + C_) + (unsigned)lane * 4u;

  // Extended labels: ext[even]=BLANK, ext[2j+1]=label[j].
  // skip[s] legal iff s odd, s>=3 and label[j]!=label[j-1].
  const bool odd = (lane & 1) != 0;
  int  e0 = BLANK_, e1 = BLANK_;
  bool sk0 = false, sk1 = false;
  if (odd) {
    const int j0 = (lane - 1) >> 1;  // 0..15  (s = lane)
    const int j1 = j0 + 16;          // 16..31 (s = lane+32)
    e0  = yt[j0];
    e1  = yt[j1];
    sk0 = (j0 >= 1) && (e0 != yt[j0 - 1]);
    sk1 = (e1 != yt[j1 - 1]);
  }

  // Prefetch t=0 and t=1 slices; wait until t=0 (first 3 per lane) resident.
  async_load3(lds0, (unsigned)lane * 4u, gbase);
  async_load3(lds1, (unsigned)(C_ + lane) * 4u, gbase);
  wait_async_le3();

  // ---- t = 0: alpha init ----
  float a0, a1, a2;
  {
    const float* b0 = buf;
    float lse = lse_row(b0, lane);
    float ainit = b0[e0] - lse;            // lp_ext[0, s] for s = lane
    a0 = (lane < 2) ? ainit : NEGF;        // alpha0 set only at s=0,1
    a1 = NEGF;
    a2 = NEGF;
  }

  // ---- forward recursion over time ----
  for (int t = 1; t < T_; ++t) {
    if (t + 1 < T_) {
      const unsigned goff = (unsigned)((unsigned)(t + 1) * C_ + lane) * 4u;
      async_load3(((t + 1) & 1) ? lds1 : lds0, goff, gbase);
      wait_async_le3();                    // slice t done; t+1 in flight
    } else {
      wait_async_0();                      // last slice: drain fully
    }

    const float* bt = buf + (t & 1) * C_;
    const float  lse = lse_row(bt, lane);
    const float  lp0 = bt[e0]     - lse;   // lp_ext[t, lane]
    const float  lp1 = bt[e1]     - lse;   // lp_ext[t, lane+32]
    const float  lp2 = bt[BLANK_] - lse;   // lp_ext[t, 64] (blank)

    // cross-lane boundary values (capture before overwrite)
    const float a0_31 = __shfl(a0, 31, 32);   // alpha[31]
    const float a1_31 = __shfl(a1, 31, 32);   // alpha[63]

    // alpha[s-1] / alpha[s-2] for s = lane
    float p0 = __shfl_up(a0, 1, 32); if (lane == 0) p0 = NEGF;
    float q0 = __shfl_up(a0, 2, 32); q0 = (lane >= 2 && sk0) ? q0 : NEGF;

    // alpha[s-1] / alpha[s-2] for s = lane+32
    float p1 = __shfl_up(a1, 1, 32); if (lane == 0) p1 = a0_31;      // alpha[31]
    float q1r = __shfl_up(a1, 2, 32);
    float q1 = NEGF;
    if (lane >= 2)      q1 = sk1 ? q1r   : NEGF;
    else if (lane == 1) q1 = sk1 ? a0_31 : NEGF;                     // s=33 -> alpha[31]
    // lane 0: s=32 is a blank -> no skip

    const float n0 = lse3(a0, p0, q0) + lp0;
    const float n1 = lse3(a1, p1, q1) + lp1;
    const float n2 = lse3(a2, a1_31, NEGF) + lp2;  // s=64: stay + alpha[63]

    a0 = n0; a1 = n1; a2 = n2;
  }

  // loss = -logsumexp(alpha_T[S-1], alpha_T[S-2])
  const float a63 = __shfl(a1, 31, 32);
  if (lane == 0) {
    float m = fmaxf(a2, a63);
    float l = m + __logf(__expf(a2 - m) + __expf(a63 - m));
    loss[b] = -l;
  }
}

extern "C" void kernel_launch(void* const* d_in, const int* in_sizes, int n_in,
                              void* d_out, int out_size, void* d_ws, size_t ws_size,
                              hipStream_t stream) {
  (void)in_sizes; (void)n_in; (void)out_size; (void)d_ws; (void)ws_size;
  const int*   y_true = (const int*)d_in[0];    // labels [B, L]
  const float* y_pred = (const float*)d_in[1];  // logits [B, T, C]
  float*       out    = (float*)d_out;          // loss   [B, 1]
  ctc_fwd_kernel<<<B_, 32, 0, stream>>>(y_true, y_pred, out);
}